// DKTAccum_no_tempo_Model_41927470744185
// MI455X (gfx1250) — compile-verified
//
#include <hip/hip_runtime.h>
#include <hip/hip_bf16.h>
#include <math.h>

typedef __attribute__((ext_vector_type(2))) float v2f;
typedef __attribute__((ext_vector_type(8))) float v8f;

#define B_ 128
#define T_ 500
#define S_ 200
#define E_ 100
#define H_ 100
#define G4 400   // 4*H
#define X2S 400  // 2*S

__device__ __forceinline__ float sigmoidf_(float x) {
    return 1.0f / (1.0f + expf(-x));
}

// -----------------------------------------------------------------------------
// Kernel 1: v[n] = lstm_b[n] + sum_k bx[k]*lstm_k[k][n]   (400 outputs)
// -----------------------------------------------------------------------------
__global__ void bias_v_kernel(const float* __restrict__ bx,
                              const float* __restrict__ lstm_k,
                              const float* __restrict__ lstm_b,
                              float* __restrict__ v) {
    int n = blockIdx.x * blockDim.x + threadIdx.x;
    if (n < G4) {
        float s = lstm_b[n];
        #pragma unroll 4
        for (int k = 0; k < E_; ++k) s += bx[k] * lstm_k[k * G4 + n];
        v[n] = s;
    }
}

// -----------------------------------------------------------------------------
// Kernel 2: G[r][n] = sum_k Wx[r][k]*lstm_k[k][n] + v[n]   (400x400)
// One wave per 16x16 tile, f32 WMMA, K=100 = 25 x K4.
// -----------------------------------------------------------------------------
__global__ __launch_bounds__(32) void gemm_g_kernel(const float* __restrict__ Wx,
                                                    const float* __restrict__ lstm_k,
                                                    const float* __restrict__ v,
                                                    float* __restrict__ Gp) {
    int lane = threadIdx.x & 31;
    int ti = blockIdx.x;          // M tile 0..24
    int tj = blockIdx.y;          // N tile 0..24
    int l15 = lane & 15;
    int half = (lane < 16) ? 0 : 1;
    int n = tj * 16 + l15;

    v8f acc;
    float vv = v[n];
    #pragma unroll
    for (int r = 0; r < 8; ++r) acc[r] = vv;

    int row = ti * 16 + l15;       // A: M = lane&15 in both halves
    int kBase = half * 2;          // A/B: VGPR0 holds K=0 (lo half) / K=2 (hi half)
    #pragma unroll
    for (int kk = 0; kk < 25; ++kk) {
        int k = kk * 4 + kBase;
        v2f a;  a.x = Wx[row * E_ + k];       a.y = Wx[row * E_ + k + 1];
        v2f bb; bb.x = lstm_k[k * G4 + n];    bb.y = lstm_k[(k + 1) * G4 + n];
        acc = __builtin_amdgcn_wmma_f32_16x16x4_f32(false, a, false, bb,
                                                    (short)0, acc, false, false);
    }
    #pragma unroll
    for (int r = 0; r < 8; ++r) {
        int m = r + half * 8;      // C/D: lanes 0-15 -> M=r, lanes 16-31 -> M=r+8
        Gp[(ti * 16 + m) * G4 + n] = acc[r];
    }
}

// -----------------------------------------------------------------------------
// Kernel 3: per-batch prep. argmax over x (400) and q (200) per t, then a
// serial cumulative-count scan producing cc/ic = log1p(counts) features.
// One block per batch element.
// -----------------------------------------------------------------------------
__global__ __launch_bounds__(256) void prep_kernel(const float* __restrict__ x,
                                                   const float* __restrict__ q,
                                                   int* __restrict__ idx2,
                                                   int* __restrict__ qiO,
                                                   float* __restrict__ ccO,
                                                   float* __restrict__ icO) {
    __shared__ int sIdx[T_];
    __shared__ int cnt[X2S];
    int b = blockIdx.x;
    int tid = threadIdx.x, lane = tid & 31, wave = tid >> 5;

    for (int t = wave; t < T_; t += 8) {
        const float* xr = x + ((size_t)b * T_ + t) * X2S;
        float bv = -3.4e38f; int bi = 1 << 30;
        for (int n = lane; n < X2S; n += 32) {
            float vv = xr[n];
            if (vv > bv || (vv == bv && n < bi)) { bv = vv; bi = n; }
        }
        for (int off = 16; off; off >>= 1) {
            float ov = __shfl_xor(bv, off, 32);
            int   oi = __shfl_xor(bi, off, 32);
            if (ov > bv || (ov == bv && oi < bi)) { bv = ov; bi = oi; }
        }
        int id = (bv != 0.0f) ? bi : -1;
        if (lane == 0) { sIdx[t] = id; idx2[(size_t)b * T_ + t] = id; }

        const float* qr = q + ((size_t)b * T_ + t) * S_;
        bv = -3.4e38f; bi = 1 << 30;
        for (int n = lane; n < S_; n += 32) {
            float vv = qr[n];
            if (vv > bv || (vv == bv && n < bi)) { bv = vv; bi = n; }
        }
        for (int off = 16; off; off >>= 1) {
            float ov = __shfl_xor(bv, off, 32);
            int   oi = __shfl_xor(bi, off, 32);
            if (ov > bv || (ov == bv && oi < bi)) { bv = ov; bi = oi; }
        }
        if (lane == 0) qiO[(size_t)b * T_ + t] = (bv != 0.0f) ? bi : -1;
    }

    for (int n = tid; n < X2S; n += 256) cnt[n] = 0;
    __syncthreads();

    if (tid == 0) {
        for (int t = 0; t < T_; ++t) {
            int id = sIdx[t];
            float ccv = 0.f, icv = 0.f;
            if (id >= 0) {
                cnt[id] += 1;                       // counts include current step
                int sk = id >> 1;
                ccv = log1pf((float)cnt[2 * sk]);
                icv = log1pf((float)cnt[2 * sk + 1]);
            }
            ccO[(size_t)b * T_ + t] = ccv;
            icO[(size_t)b * T_ + t] = icv;
        }
    }
}

// -----------------------------------------------------------------------------
// Kernel 4: LSTM recurrence. 8 blocks x 16 batch rows, 16 waves.
//  - Rk lives in LDS for all 500 steps, swizzled [kk][n][4] so each B fragment
//    is one conflict-free ds_load_b64.
//  - A fragments (h tile) hoisted into 25 v2f registers per wave per step.
//  - z-init (G[idx] + count features) staged cooperatively into sZ.
//  - If hseq != nullptr, h is streamed out and the output dot is deferred to
//    out_kernel; otherwise it is computed in-loop (fallback).
// -----------------------------------------------------------------------------
__global__ __launch_bounds__(512) void lstm_kernel(const float* __restrict__ Gp,
                                                   const float* __restrict__ lstm_k,
                                                   const float* __restrict__ Rk,
                                                   const float* __restrict__ Wo,
                                                   const float* __restrict__ bo,
                                                   const int* __restrict__ idx2,
                                                   const int* __restrict__ qiI,
                                                   const float* __restrict__ cc,
                                                   const float* __restrict__ ic,
                                                   float* __restrict__ hseq,
                                                   float* __restrict__ out) {
    __shared__ float sRk[H_ * G4];     // 160000 B, swizzled [kk][n][4]
    __shared__ float sH[16 * H_];      // 6400 B
    __shared__ float sC[16 * H_];      // 6400 B
    __shared__ float sZ[16 * G4];      // 25600 B
    __shared__ float sKA[G4], sKB[G4], sKC[G4];
    __shared__ int   sIdx[16];
    __shared__ int   sQi[16];

    int tid = threadIdx.x, lane = tid & 31, wave = tid >> 5;   // 16 waves
    int bBase = blockIdx.x * 16;
    int l15 = lane & 15;
    int half = (lane < 16) ? 0 : 1;
    int kBase = half * 2;

    // load Rk swizzled: sRk[(kk*G4 + n)*4 + (k&3)] = Rk[k*G4 + n]
    for (int i = tid; i < H_ * G4; i += 512) {
        int k = i / G4, n = i - k * G4;
        sRk[((k >> 2) * G4 + n) * 4 + (k & 3)] = Rk[i];
    }
    for (int n = tid; n < G4; n += 512) {
        sKA[n] = lstm_k[(E_ + 0) * G4 + n];
        sKB[n] = lstm_k[(E_ + 1) * G4 + n];
        sKC[n] = lstm_k[(E_ + 2) * G4 + n];
    }
    for (int i = tid; i < 16 * H_; i += 512) { sH[i] = 0.f; sC[i] = 0.f; }
    __syncthreads();

    for (int t = 0; t < T_; ++t) {
        // ---- stage z-init: wave w handles batch row m = w -------------------
        {
            int m = wave;
            size_t o = (size_t)(bBase + m) * T_ + t;
            int ix = idx2[o];                       // broadcast load (uniform addr)
            float cv = cc[o], iv = ic[o];
            if (lane == 0) {
                sIdx[m] = ix;
                if (hseq == nullptr) sQi[m] = qiI[o];
            }
            int g = (ix < 0) ? 0 : ix;              // masked rows: value unused
            const float* grow = Gp + (size_t)g * G4;
            for (int j = lane; j < G4; j += 32) {
                sZ[m * G4 + j] = grow[j] + cv * sKA[j] + iv * sKB[j]
                                 + (cv + iv) * sKC[j];
            }
        }
        __syncthreads();

        // ---- z += h @ Rk via f32 WMMA --------------------------------------
        {
            // hoist A fragments: same h tile for every N-tile of this wave
            int aoff = l15 * H_ + kBase;
            v2f areg[25];
            #pragma unroll
            for (int kk = 0; kk < 25; ++kk) {
                areg[kk].x = sH[aoff + kk * 4];
                areg[kk].y = sH[aoff + kk * 4 + 1];
            }
            for (int nt = wave; nt < 25; nt += 16) {
                int n = nt * 16 + l15;
                v8f acc;
                #pragma unroll
                for (int r = 0; r < 8; ++r)
                    acc[r] = sZ[(r + half * 8) * G4 + n];
                int bOff = n * 4 + kBase;           // element offset in sRk slice
                #pragma unroll
                for (int kk = 0; kk < 25; ++kk) {
                    const v2f* bp = reinterpret_cast<const v2f*>(sRk + kk * (G4 * 4) + bOff);
                    acc = __builtin_amdgcn_wmma_f32_16x16x4_f32(false, areg[kk],
                            false, *bp, (short)0, acc, false, false);
                }
                #pragma unroll
                for (int r = 0; r < 8; ++r)
                    sZ[(r + half * 8) * G4 + n] = acc[r];
            }
        }
        __syncthreads();

        // ---- gates + state update (+ h_seq stream-out) ---------------------
        for (int p = tid; p < 16 * H_; p += 512) {
            int m = p / H_, j = p - m * H_;
            float zi = sZ[m * G4 + j];
            float zf = sZ[m * G4 + H_ + j];
            float zg = sZ[m * G4 + 2 * H_ + j];
            float zo = sZ[m * G4 + 3 * H_ + j];
            float cOld = sC[m * H_ + j];
            float hOld = sH[m * H_ + j];
            float cn = sigmoidf_(zf) * cOld + sigmoidf_(zi) * tanhf(zg);
            float hn = sigmoidf_(zo) * tanhf(cn);
            bool upd = (sIdx[m] >= 0);
            float cf = upd ? cn : cOld;
            float hf = upd ? hn : hOld;
            sC[m * H_ + j] = cf;
            sH[m * H_ + j] = hf;
            if (hseq != nullptr)
                hseq[((size_t)(bBase + m) * T_ + t) * H_ + j] = hf;
        }
        __syncthreads();

        // ---- fallback: fused output dot (wave w -> row w) ------------------
        if (hseq == nullptr) {
            int m = wave;
            int qv = sQi[m];
            float sum = 0.f;
            if (qv >= 0) {
                for (int k = lane; k < H_; k += 32)
                    sum += sH[m * H_ + k] * Wo[k * S_ + qv];
            }
            for (int off = 16; off; off >>= 1) sum += __shfl_xor(sum, off, 32);
            if (lane == 0) {
                out[(size_t)(bBase + m) * T_ + t] =
                    (qv >= 0) ? sigmoidf_(sum + bo[qv]) : 0.f;
            }
            __syncthreads();
        }
    }
}

// -----------------------------------------------------------------------------
// Kernel 5 (deferred-output mode): out[bt] = sigmoid(h_seq[bt]·Wo[:,qi]+bo[qi])
// One wave per (b,t) pair.
// -----------------------------------------------------------------------------
__global__ __launch_bounds__(256) void out_kernel(const float* __restrict__ hseq,
                                                  const float* __restrict__ Wo,
                                                  const float* __restrict__ bo,
                                                  const int* __restrict__ qiI,
                                                  float* __restrict__ out) {
    int wave = threadIdx.x >> 5, lane = threadIdx.x & 31;
    int bt = blockIdx.x * 8 + wave;
    if (bt >= B_ * T_) return;
    int qv = qiI[bt];
    float sum = 0.f;
    if (qv >= 0) {
        const float* hr = hseq + (size_t)bt * H_;
        for (int k = lane; k < H_; k += 32) sum += hr[k] * Wo[k * S_ + qv];
    }
    for (int off = 16; off; off >>= 1) sum += __shfl_xor(sum, off, 32);
    if (lane == 0) out[bt] = (qv >= 0) ? sigmoidf_(sum + bo[qv]) : 0.f;
}

// -----------------------------------------------------------------------------
extern "C" void kernel_launch(void* const* d_in, const int* in_sizes, int n_in,
                              void* d_out, int out_size, void* d_ws, size_t ws_size,
                              hipStream_t stream) {
    (void)in_sizes; (void)n_in; (void)out_size;
    const float* x       = (const float*)d_in[0];
    // d_in[1] = delta, unused by the reference forward pass
    const float* q       = (const float*)d_in[2];
    const float* Wx      = (const float*)d_in[3];
    const float* bx      = (const float*)d_in[4];
    const float* lstm_k  = (const float*)d_in[5];
    const float* lstm_rk = (const float*)d_in[6];
    const float* lstm_b  = (const float*)d_in[7];
    const float* Wo      = (const float*)d_in[8];
    const float* bo      = (const float*)d_in[9];
    float* out = (float*)d_out;

    char* ws = (char*)d_ws;
    float* Gp   = (float*)(ws + 0);        // 400*400*4 = 640000
    float* v    = (float*)(ws + 640000);   // 1600
    int*   idx  = (int*)  (ws + 641600);   // 256000
    int*   qiB  = (int*)  (ws + 897600);   // 256000
    float* ccB  = (float*)(ws + 1153600);  // 256000
    float* icB  = (float*)(ws + 1409600);  // 256000
    const size_t HSEQ_OFF = 1665600;       // B*T*H*4 = 25600000
    bool deferred = (ws_size >= HSEQ_OFF + (size_t)B_ * T_ * H_ * 4);
    float* hseq = deferred ? (float*)(ws + HSEQ_OFF) : nullptr;

    bias_v_kernel<<<2, 256, 0, stream>>>(bx, lstm_k, lstm_b, v);
    gemm_g_kernel<<<dim3(25, 25), 32, 0, stream>>>(Wx, lstm_k, v, Gp);
    prep_kernel<<<B_, 256, 0, stream>>>(x, q, idx, qiB, ccB, icB);
    lstm_kernel<<<8, 512, 0, stream>>>(Gp, lstm_k, lstm_rk, Wo, bo,
                                       idx, qiB, ccB, icB, hseq, out);
    if (deferred) {
        out_kernel<<<(B_ * T_ + 7) / 8, 256, 0, stream>>>(hseq, Wo, bo, qiB, out);
    }
}